// BallQuery_43714177139075
// MI455X (gfx1250) — compile-verified
//
#include <hip/hip_runtime.h>

// Ball query (PointNet++) on MI455X / gfx1250.
// Phase 1: squared distances via V_WMMA_F32_16X16X4_F32:
//   A[m] = (-2cx, -2cy, -2cz, 1), B[n] = (px, py, pz, p2), C[m][*] = c2[m]
//   => D[m][n] = |p_n - c_m|^2 in a single WMMA per 16x16 tile.
// Phase 2: merge per-chunk first-K index lists + gather coords/features.

#define BQ_B 4
#define BQ_N 16384
#define BQ_M 2048
#define BQ_C 64
#define BQ_K 32
#define BQ_R2 0.01f

typedef __attribute__((ext_vector_type(2))) float v2f;
typedef __attribute__((ext_vector_type(8))) float v8f;

__device__ __forceinline__ v8f wmma_dist2(v2f a, v2f b, v8f c) {
    // 8-arg form: (neg_a, A, neg_b, B, c_mod, C, reuse_a, reuse_b)
    return __builtin_amdgcn_wmma_f32_16x16x4_f32(false, a, false, b,
                                                 (short)0, c, false, false);
}

// ---------------------------------------------------------------------------
// Phase 1: each wave handles 32 centers (two 16-row WMMA tiles) over one
// N-chunk; emits per-chunk ascending index lists (capped at K) + counts.
// ---------------------------------------------------------------------------
__global__ void bq_phase1(const float* __restrict__ pc,   // (B,3,N)
                          const float* __restrict__ cc,   // (B,3,M)
                          int* __restrict__ wcnt,         // (B*M, nch)
                          int* __restrict__ widx,         // (B*M, nch, K)
                          int nch) {
    const int lane = threadIdx.x & 31;
    const int wid  = threadIdx.x >> 5;
    const int gw   = blockIdx.x * (blockDim.x >> 5) + wid;
    const int mtiles = BQ_M / 32;
    const int ch = gw % nch;
    const int mt = (gw / nch) % mtiles;
    const int b  = gw / (nch * mtiles);
    if (b >= BQ_B) return;

    const int m0      = mt * 32;
    const int chn     = BQ_N / nch;
    const int n_begin = ch * chn;
    const int n_end   = n_begin + chn;
    const bool hiHalf = (lane >= 16);

    // ---- A matrices (16x4 fp32 layout: lanes 0-15 hold K=0,1; 16-31 hold K=2,3)
    const float* ccb = cc + b * 3 * BQ_M;
    const int mlo = m0 + (lane & 15);
    const int mhi = mlo + 16;
    const float cx0 = ccb[0 * BQ_M + mlo], cy0 = ccb[1 * BQ_M + mlo], cz0 = ccb[2 * BQ_M + mlo];
    const float cx1 = ccb[0 * BQ_M + mhi], cy1 = ccb[1 * BQ_M + mhi], cz1 = ccb[2 * BQ_M + mhi];

    v2f a_lo, a_hi;
    a_lo.x = hiHalf ? (-2.0f * cz0) : (-2.0f * cx0);
    a_lo.y = hiHalf ? 1.0f          : (-2.0f * cy0);
    a_hi.x = hiHalf ? (-2.0f * cz1) : (-2.0f * cx1);
    a_hi.y = hiHalf ? 1.0f          : (-2.0f * cy1);

    // c2 of "my" center (m0 + lane); distribute per C/D row via shuffles.
    const float myc2 = hiHalf ? (cx1 * cx1 + cy1 * cy1 + cz1 * cz1)
                              : (cx0 * cx0 + cy0 * cy0 + cz0 * cz0);
    v8f c_lo, c_hi;
    const int rbase = hiHalf ? 8 : 0;
#pragma unroll
    for (int r = 0; r < 8; ++r) {
        c_lo[r] = __shfl(myc2, rbase + r, 32);        // row r(+8) of tile m0..m0+15
        c_hi[r] = __shfl(myc2, 16 + rbase + r, 32);   // row r(+8) of tile m0+16..m0+31
    }

    const float* pcb = pc + b * 3 * BQ_N;
    const int myCenter = b * BQ_M + m0 + lane;        // lane L owns center m0+L
    int* __restrict__ myIdx = widx + ((size_t)myCenter * nch + ch) * BQ_K;
    int cnt = 0;

    for (int n0 = n_begin; n0 < n_end; n0 += 16) {
        const int np = n0 + (lane & 15);
        const float px = pcb[0 * BQ_N + np];
        const float py = pcb[1 * BQ_N + np];
        const float pz = pcb[2 * BQ_N + np];
        const float p2 = px * px + py * py + pz * pz;

        v2f bm;                                        // B 4x16: lanes<16 K=0,1; >=16 K=2,3
        bm.x = hiHalf ? pz : px;
        bm.y = hiHalf ? p2 : py;

        const v8f d_lo = wmma_dist2(a_lo, bm, c_lo);   // dist2 rows m0..m0+15
        const v8f d_hi = wmma_dist2(a_hi, bm, c_hi);   // dist2 rows m0+16..m0+31

        unsigned mk_lo[8], mk_hi[8];
#pragma unroll
        for (int r = 0; r < 8; ++r) {
            mk_lo[r] = __builtin_amdgcn_ballot_w32(d_lo[r] < BQ_R2);
            mk_hi[r] = __builtin_amdgcn_ballot_w32(d_hi[r] < BQ_R2);
        }
        // Redistribute: lane L gets the 16-bit column mask of its own center row.
        const int rsel = lane & 7;
        unsigned msel = hiHalf ? mk_hi[0] : mk_lo[0];
#pragma unroll
        for (int r = 1; r < 8; ++r) {
            const unsigned cand = hiHalf ? mk_hi[r] : mk_lo[r];
            msel = (rsel == r) ? cand : msel;
        }
        unsigned bits = ((lane & 15) < 8) ? (msel & 0xffffu) : (msel >> 16);

        while (bits) {                                  // ~0.07 set bits expected/tile
            if (cnt >= BQ_K) break;
            const int t = __builtin_ctz(bits);
            bits &= bits - 1;
            myIdx[cnt++] = n0 + t;
        }
        if (__builtin_amdgcn_ballot_w32(cnt < BQ_K) == 0) break;  // all 32 centers full
    }
    wcnt[(size_t)myCenter * nch + ch] = cnt;
}

// ---------------------------------------------------------------------------
// Phase 2: one wave per center; lane k resolves the global k-th neighbor by
// prefix-walking the per-chunk counts, then streams 67 channels (coalesced).
// ---------------------------------------------------------------------------
__global__ void bq_phase2(const float* __restrict__ pc, const float* __restrict__ cc,
                          const float* __restrict__ pf, const int* __restrict__ wcnt,
                          const int* __restrict__ widx, float* __restrict__ out, int nch) {
    const int lane = threadIdx.x & 31;
    const int wid  = threadIdx.x >> 5;
    const int cidx = blockIdx.x * (blockDim.x >> 5) + wid;
    if (cidx >= BQ_B * BQ_M) return;
    const int b = cidx / BQ_M;
    const int m = cidx - b * BQ_M;
    const int k = lane;

    int pre = 0, nidx = -1, firstIdx = -1;
    for (int c = 0; c < nch; ++c) {
        const int cn = wcnt[(size_t)cidx * nch + c];
        const int base = ((size_t)cidx * nch + c) * BQ_K;
        if (cn > 0 && firstIdx < 0) firstIdx = widx[base];
        if (nidx < 0 && k >= pre && k < pre + cn) nidx = widx[base + (k - pre)];
        pre += cn;
    }
    if (nidx < 0) nidx = (firstIdx >= 0) ? firstIdx : 0;   // pad w/ first hit, else 0

    const float* pcb = pc + b * 3 * BQ_N;
    const float* ccb = cc + b * 3 * BQ_M;
    const float* pfb = pf + (size_t)b * BQ_C * BQ_N;
    float* ob = out + (size_t)b * (3 + BQ_C) * BQ_M * BQ_K + (size_t)m * BQ_K + k;

#pragma unroll
    for (int c = 0; c < 3; ++c)                             // centered coords
        ob[(size_t)c * BQ_M * BQ_K] = pcb[c * BQ_N + nidx] - ccb[c * BQ_M + m];
    for (int c = 0; c < BQ_C; ++c)                          // features (L2-resident gather)
        ob[(size_t)(3 + c) * BQ_M * BQ_K] = pfb[(size_t)c * BQ_N + nidx];
}

extern "C" void kernel_launch(void* const* d_in, const int* in_sizes, int n_in,
                              void* d_out, int out_size, void* d_ws, size_t ws_size,
                              hipStream_t stream) {
    (void)in_sizes; (void)n_in; (void)out_size;
    const float* pc = (const float*)d_in[0];   // points_coords  (4,3,16384)
    const float* cc = (const float*)d_in[1];   // centers_coords (4,3,2048)
    const float* pf = (const float*)d_in[2];   // points_features(4,64,16384)
    float* out = (float*)d_out;                // (4,67,2048,32)

    // Pick chunk count from available scratch (counts + index lists).
    int nch = 8;
    while (nch > 1 &&
           (size_t)BQ_B * BQ_M * nch * (BQ_K + 1) * sizeof(int) > ws_size)
        nch >>= 1;

    int* wcnt = (int*)d_ws;
    int* widx = wcnt + (size_t)BQ_B * BQ_M * nch;

    const int wavesPerBlk = 256 / 32;
    const int waves1 = BQ_B * (BQ_M / 32) * nch;
    bq_phase1<<<(waves1 + wavesPerBlk - 1) / wavesPerBlk, 256, 0, stream>>>(
        pc, cc, wcnt, widx, nch);

    const int waves2 = BQ_B * BQ_M;
    bq_phase2<<<(waves2 + wavesPerBlk - 1) / wavesPerBlk, 256, 0, stream>>>(
        pc, cc, pf, wcnt, widx, out, nch);
}